// RoiAlign_77111842832905
// MI455X (gfx1250) — compile-verified
//
#include <hip/hip_runtime.h>
#include <hip/hip_bf16.h>
#include <stdint.h>

typedef __attribute__((ext_vector_type(2))) float v2f;
typedef __attribute__((ext_vector_type(8))) float v8f;

#define CROP     14
#define CH       256
#define ROWS     56      // 4 corners * 14 grid-x positions
#define LDS_ROW  272     // padded row stride (floats): kills 2-way bank conflicts
#define IMG_F    1024.0f

// One block per (box, grid-row gy). 8 wave32s; wave w owns channel tiles {w, w+8}.
// Phase 1: async-stage the 56 gathered 1KB feature rows into LDS (ASYNCcnt path).
// Phase 2: out[14x256] = A[14x56] x G[56x256] on the matrix pipe (f32 WMMA, K=4 steps).
__global__ __launch_bounds__(256)
void roialign_fpn_wmma(const float* __restrict__ boxes,
                       const float* __restrict__ f0, const float* __restrict__ f1,
                       const float* __restrict__ f2, const float* __restrict__ f3,
                       const float* __restrict__ f4,
                       float* __restrict__ out, int N)
{
    __shared__ float smem[ROWS * LDS_ROW];   // 59.5 KB

    const int tid  = threadIdx.x;
    const int lane = tid & 31;
    const int m    = lane & 15;            // M (grid-x) for A, N (channel) for B/C
    const bool hi  = lane >= 16;
    const int wv   = tid >> 5;             // wave 0..7

    const int gy  = blockIdx.x % CROP;
    const int box = blockIdx.x / CROP;
    const int b   = box / N;

    // ---- box params (uniform per block; recomputed per thread, trivially cheap) ----
    const float x1 = boxes[box * 4 + 0];
    const float y1 = boxes[box * 4 + 1];
    const float x2 = boxes[box * 4 + 2];
    const float y2 = boxes[box * 4 + 3];
    const float bw = x2 - x1, bh = y2 - y1;
    int level = (int)floorf(1.0f + log2f(sqrtf(bw * bh) / 224.0f + 1e-7f));
    level = level < 0 ? 0 : (level > 4 ? 4 : level);
    const int   dim   = 128 >> level;            // feature H == W
    const float scale = (float)dim / IMG_F;      // 1/stride
    const float fmax  = (float)(dim - 1);
    const float* feat = (level == 0) ? f0 : (level == 1) ? f1 :
                        (level == 2) ? f2 : (level == 3) ? f3 : f4;

    // y sampling for this grid row:  ys = y1*scale + gy*((bh*scale - 1)/13)
    const float ys  = y1 * scale + (float)gy * ((bh * scale - 1.0f) * (1.0f / 13.0f));
    const bool  vy  = (ys >= 0.0f) && (ys <= fmax);
    const float ysc = fminf(fmaxf(ys, 0.0f), fmax);
    const int   iy0 = (int)floorf(ysc);
    const int   iyc = (int)ceilf(ysc);
    const float ly  = ysc - floorf(ysc);

    const float sx = x1 * scale;
    const float dx = (bw * scale - 1.0f) * (1.0f / 13.0f);

    // ---- Phase 1: stage 56 rows x 256 ch = 3584 float4 transfers (14 / thread) ----
#pragma unroll
    for (int j = 0; j < 14; ++j) {
        const int id    = j * 256 + tid;
        const int row   = id >> 6;         // 0..55
        const int chunk = id & 63;         // float4 within the channel row
        const int xp    = row >> 2;        // grid-x 0..13
        const int cr    = row & 3;         // corner: 00,0c,c0,cc
        const float xs  = sx + (float)xp * dx;
        const float xsc = fminf(fmaxf(xs, 0.0f), fmax);
        const int ix0 = (int)floorf(xsc);
        const int ixc = (int)ceilf(xsc);
        const int yy  = (cr < 2) ? iy0 : iyc;
        const int xx  = (cr & 1) ? ixc : ix0;
        const float* src = feat + (((size_t)b * dim + yy) * dim + xx) * CH + chunk * 4;
        const unsigned ldsa = (unsigned)(uintptr_t)(&smem[row * LDS_ROW + chunk * 4]);
        asm volatile("global_load_async_to_lds_b128 %0, %1, off"
                     :: "v"(ldsa), "v"(src) : "memory");
    }
    __builtin_amdgcn_s_wait_asynccnt(0);   // async copies are NOT covered by the barrier
    __syncthreads();

    // ---- Phase 2: D = A x G on the matrix pipe ----
    // A[16x56]: row m has its 4 bilinear weights at columns 4m..4m+3 (mask folded in).
    // k-step s touches columns 4s..4s+3  =>  A nonzero only in row m == s.
    v8f acc0 = {};
    v8f acc1 = {};
    const int c0    = wv * 16 + m;          // channel tile wv
    const int c1    = (wv + 8) * 16 + m;    // channel tile wv+8
    const int rsel  = hi ? 2 : 0;           // lane-half K selection (K=0/1 vs 2/3)

    for (int s = 0; s < CROP; ++s) {
        const float xs  = sx + (float)s * dx;
        const bool  vx  = (xs >= 0.0f) && (xs <= fmax);
        const float xsc = fminf(fmaxf(xs, 0.0f), fmax);
        const float lx  = xsc - floorf(xsc);
        const float msk = (vx && vy) ? 1.0f : 0.0f;
        const float w0  = msk * (1.0f - ly) * (1.0f - lx);
        const float w1  = msk * (1.0f - ly) * lx;
        const float w2  = msk * ly * (1.0f - lx);
        const float w3  = msk * ly * lx;

        v2f a;
        a.x = (m == s) ? (hi ? w2 : w0) : 0.0f;
        a.y = (m == s) ? (hi ? w3 : w1) : 0.0f;

        const int rbase = (4 * s + rsel) * LDS_ROW;
        v2f b0, b1;
        b0.x = smem[rbase + c0];
        b0.y = smem[rbase + LDS_ROW + c0];
        b1.x = smem[rbase + c1];
        b1.y = smem[rbase + LDS_ROW + c1];

        acc0 = __builtin_amdgcn_wmma_f32_16x16x4_f32(false, a, false, b0,
                                                     (short)0, acc0, false, false);
        acc1 = __builtin_amdgcn_wmma_f32_16x16x4_f32(false, a, false, b1,
                                                     (short)0, acc1, false, false);
    }

    // ---- store: C VGPR r holds M=r (lanes 0-15) / M=r+8 (lanes 16-31) ----
    float* obase = out + ((size_t)box * CROP + gy) * CROP * CH;
#pragma unroll
    for (int r = 0; r < 8; ++r) {
        const int xr = hi ? (r + 8) : r;
        if (xr < CROP) {
            obase[(size_t)xr * CH + c0] = acc0[r];
            obase[(size_t)xr * CH + c1] = acc1[r];
        }
    }
}

extern "C" void kernel_launch(void* const* d_in, const int* in_sizes, int n_in,
                              void* d_out, int out_size, void* d_ws, size_t ws_size,
                              hipStream_t stream) {
    // setup_inputs order: image_shape, boxes, scores, fpn0..fpn4
    const float* boxes = (const float*)d_in[1];
    const float* f0 = (const float*)d_in[3];
    const float* f1 = (const float*)d_in[4];
    const float* f2 = (const float*)d_in[5];
    const float* f3 = (const float*)d_in[6];
    const float* f4 = (const float*)d_in[7];
    float* out = (float*)d_out;

    const int N  = 300;
    const int BN = in_sizes[2];            // scores: B*N
    dim3 grid(BN * CROP), block(256);
    roialign_fpn_wmma<<<grid, block, 0, stream>>>(boxes, f0, f1, f2, f3, f4, out, N);
}